// PresetEmbedding_8572754723047
// MI455X (gfx1250) — compile-verified
//
#include <hip/hip_runtime.h>
#include <stdint.h>

// Problem constants (fixed by the reference)
#define LPAR 256
#define HDIM 64
#define NTYP 16
#define MAXC 32

typedef float v4f __attribute__((ext_vector_type(4)));
typedef int   v4i __attribute__((ext_vector_type(4)));

#if __has_builtin(__builtin_amdgcn_global_load_async_to_lds_b128)
#define HAVE_ASYNC_LDS 1
typedef __attribute__((address_space(1))) v4i gv4i;  // global int4
typedef __attribute__((address_space(3))) v4i lv4i;  // LDS int4
#else
#define HAVE_ASYNC_LDS 0
#endif

__device__ __forceinline__ void copy16_async(const float* __restrict__ g, float* l) {
#if HAVE_ASYNC_LDS
  __builtin_amdgcn_global_load_async_to_lds_b128((gv4i*)g, (lv4i*)l, 0, 0);
#else
  *(v4f*)l = *(const v4f*)g;
#endif
}

__device__ __forceinline__ void wait_async_all() {
#if HAVE_ASYNC_LDS
#if __has_builtin(__builtin_amdgcn_s_wait_asynccnt)
  __builtin_amdgcn_s_wait_asynccnt(0);
#else
  asm volatile("s_wait_asynccnt 0" ::: "memory");
#endif
#endif
}

__global__ __launch_bounds__(256) void preset_embedding_kernel(
    const float* __restrict__ u_in,      // (N, L, 3)
    const float* __restrict__ conv_w,    // (NT*H,)
    const float* __restrict__ conv_b,    // (NT*H,)
    const float* __restrict__ cat_table, // (MAXC*NT, H)
    float* __restrict__ out,             // (N, L, H)
    int N)
{
  // LDS-resident tables: 128KB table + 4KB w + 4KB b = 136KB (fits in 320KB WGP LDS)
  __shared__ float s_tab[MAXC * NTYP * HDIM]; // 512*64
  __shared__ float s_w[NTYP * HDIM];          // 1024
  __shared__ float s_b[NTYP * HDIM];          // 1024

  const int tid = threadIdx.x;

  // ---- Stage tables into LDS via async global->LDS (ASYNCcnt path) ----
  // cat_table: 8192 float4 chunks, 32 per thread
  for (int i = tid; i < (MAXC * NTYP * HDIM) / 4; i += 256)
    copy16_async(cat_table + 4 * i, s_tab + 4 * i);
  // conv_w / conv_b: 256 float4 chunks each, 1 per thread
  for (int i = tid; i < (NTYP * HDIM) / 4; i += 256) {
    copy16_async(conv_w + 4 * i, s_w + 4 * i);
    copy16_async(conv_b + 4 * i, s_b + 4 * i);
  }
  wait_async_all();
  __syncthreads();

  const v4f* __restrict__ s_tab4 = (const v4f*)s_tab;
  const v4f* __restrict__ s_w4   = (const v4f*)s_w;
  const v4f* __restrict__ s_b4   = (const v4f*)s_b;

  const int wave = tid >> 5;       // 0..7
  const int lane = tid & 31;       // 0..31
  const int lsub = lane >> 4;      // which l of the pair (0/1)
  const int hq   = lane & 15;      // float4 index within H (h = hq*4)

  // ---- Persistent rows: each block handles several n, 8 waves x 16 l-pairs ----
  for (int n = blockIdx.x; n < N; n += gridDim.x) {
    const size_t nbase = (size_t)n * LPAR;
    for (int p = wave; p < LPAR / 2; p += 8) {
      const int l = 2 * p + lsub;
      const float* up = u_in + (nbase + (size_t)l) * 3;
      const float cval = up[0];
      const float nval = up[1];
      const float tf   = up[2];
      const int   t    = (int)(tf + 0.5f);

      v4f r;
      if (t < NTYP / 2) {
        // numeric: r = nval * w_t + b_t  (8KB LDS, conflict-free b128 reads)
        const v4f w4 = s_w4[t * (HDIM / 4) + hq];
        const v4f b4 = s_b4[t * (HDIM / 4) + hq];
        r = nval * w4 + b4;
      } else {
        // categorical: gather 64-float row from LDS table
        const int idx = __float2int_rn(tf * (float)MAXC + cval);
        r = s_tab4[idx * (HDIM / 4) + hq];
      }
      // write-once 134MB stream: non-temporal b128 store, 512B contiguous per wave
      __builtin_nontemporal_store(r, (v4f*)(out + (nbase + (size_t)l) * HDIM) + hq);
    }
  }
}

extern "C" void kernel_launch(void* const* d_in, const int* in_sizes, int n_in,
                              void* d_out, int out_size, void* d_ws, size_t ws_size,
                              hipStream_t stream) {
  const float* u_in      = (const float*)d_in[0];
  const float* conv_w    = (const float*)d_in[1];
  const float* conv_b    = (const float*)d_in[2];
  const float* cat_table = (const float*)d_in[3];
  // d_in[4..6] (num_rows/cat_rows/num_types) are implied by u_in[:,:,2]; unused.
  float* out = (float*)d_out;

  const int N = in_sizes[0] / (LPAR * 3);
  int nBlocks = 512;
  if (N < nBlocks) nBlocks = N;

  preset_embedding_kernel<<<nBlocks, 256, 0, stream>>>(u_in, conv_w, conv_b,
                                                       cat_table, out, N);
}